// SPDTangentSpace_21328807592137
// MI455X (gfx1250) — compile-verified
//
#include <hip/hip_runtime.h>

// ---------------------------------------------------------------------------
// Batched SPD matrix log via inverse scaling-and-squaring with Newton-Schulz
// square roots: all-GEMM formulation -> V_WMMA_F32_16X16X4_F32 on gfx1250.
// One workgroup (4 wave32) per 64x64 matrix, operands in padded LDS.
// ---------------------------------------------------------------------------

#define LDA    65          // padded LDS row stride (floats) to avoid bank conflicts
#define NMAT   64
#define TRI    2080        // 64*65/2
#define NSQRT  6           // number of matrix square roots (2^6 = 64)
#define NTERMS 8           // log(I+E) series terms

typedef float v2f __attribute__((ext_vector_type(2)));
typedef float v8f __attribute__((ext_vector_type(8)));

struct SharedBlk {
  float Y[NMAT * LDA];
  float Z[NMAT * LDA];
  float T[NMAT * LDA];
  float U[NMAT * LDA];
  float red[128];
  float scale[2];          // [0] = 1/||X||_F, [1] = log ||X||_F
};

// C = alpha*(A@B) + beta*I   (64x64, all-wave cooperative, trailing barrier)
__device__ __forceinline__ void wg_gemm64(float* __restrict__ C,
                                          const float* __restrict__ A,
                                          const float* __restrict__ B,
                                          float alpha, float beta,
                                          int lane, int wave)
{
  const int half = lane >> 4;      // 0: lanes 0-15, 1: lanes 16-31
  const int lid  = lane & 15;
  const float* Arow = A + (wave * 16 + lid) * LDA;

  v8f acc0 = {}, acc1 = {}, acc2 = {}, acc3 = {};

  for (int k0 = 0; k0 < NMAT; k0 += 4) {
    const int ka = k0 + 2 * half;  // A 16x4 fragment: vgpr v holds K = 2*half + v
    v2f a;
    a.x = Arow[ka];
    a.y = Arow[ka + 1];
    const float* B0 = B + ka * LDA + lid;
    const float* B1 = B0 + LDA;
    v2f b;
    b.x = B0[0];  b.y = B1[0];
    acc0 = __builtin_amdgcn_wmma_f32_16x16x4_f32(false, a, false, b, (short)0, acc0, false, false);
    b.x = B0[16]; b.y = B1[16];
    acc1 = __builtin_amdgcn_wmma_f32_16x16x4_f32(false, a, false, b, (short)0, acc1, false, false);
    b.x = B0[32]; b.y = B1[32];
    acc2 = __builtin_amdgcn_wmma_f32_16x16x4_f32(false, a, false, b, (short)0, acc2, false, false);
    b.x = B0[48]; b.y = B1[48];
    acc3 = __builtin_amdgcn_wmma_f32_16x16x4_f32(false, a, false, b, (short)0, acc3, false, false);
  }

  // D layout: VGPR r -> row (r + 8*half) within tile, lane&15 -> col
  const int mBase = wave * 16 + 8 * half;
#pragma unroll
  for (int r = 0; r < 8; ++r) {
    const int row = mBase + r;
    float* Crow = C + row * LDA;
    Crow[lid]      = alpha * acc0[r] + ((row == lid)      ? beta : 0.0f);
    Crow[16 + lid] = alpha * acc1[r] + ((row == 16 + lid) ? beta : 0.0f);
    Crow[32 + lid] = alpha * acc2[r] + ((row == 32 + lid) ? beta : 0.0f);
    Crow[48 + lid] = alpha * acc3[r] + ((row == 48 + lid) ? beta : 0.0f);
  }
  __syncthreads();
}

__global__ __launch_bounds__(128)
void spd_log_triu_kernel(const float* __restrict__ x, float* __restrict__ out)
{
  __shared__ SharedBlk s;
  const int tid  = threadIdx.x;
  const int lane = tid & 31;
  const int wave = tid >> 5;
  const float* X = x + (size_t)blockIdx.x * (NMAT * NMAT);

  // ---- load matrix into LDS (padded) + Frobenius-norm partial sums ----
  float fs = 0.0f;
#pragma unroll
  for (int j = 0; j < 8; ++j) {
    const int e = (j * 128 + tid) * 4;                 // 4096 floats / 128 thr
    const float4 v = *reinterpret_cast<const float4*>(X + e);
    const int r = e >> 6, c = e & 63;
    float* dst = &s.Y[r * LDA + c];
    dst[0] = v.x; dst[1] = v.y; dst[2] = v.z; dst[3] = v.w;
    fs += v.x * v.x + v.y * v.y + v.z * v.z + v.w * v.w;
  }
  s.red[tid] = fs;
  __syncthreads();
  for (int off = 64; off > 0; off >>= 1) {
    if (tid < off) s.red[tid] += s.red[tid + off];
    __syncthreads();
  }
  if (tid == 0) {
    const float c = sqrtf(s.red[0]);                   // ||X||_F >= lambda_max
    s.scale[0] = 1.0f / c;
    s.scale[1] = logf(c);
  }
  __syncthreads();
  const float inv_c = s.scale[0];
  const float log_c = s.scale[1];

  // ---- A = X / c  (eigenvalues in (0,1]) ----
  for (int i = tid; i < NMAT * NMAT; i += 128) {
    const int r = i >> 6, c = i & 63;
    s.Y[r * LDA + c] *= inv_c;
  }
  __syncthreads();

  float* Yb = s.Y;
  float* Zb = s.Z;
  float* Tb = s.T;
  float* Ub = s.U;

  // ---- s = 6 chained square roots via coupled Newton-Schulz ----
  for (int sq = 0; sq < NSQRT; ++sq) {
    for (int i = tid; i < NMAT * NMAT; i += 128) {     // Z = I
      const int r = i >> 6, c = i & 63;
      Zb[r * LDA + c] = (r == c) ? 1.0f : 0.0f;
    }
    __syncthreads();

    const int ni = (sq == 0) ? 18 : (sq == 1) ? 10 : (sq == 2) ? 8
                 : (sq == 3) ? 7  : 6;
    for (int it = 0; it < ni; ++it) {
      wg_gemm64(Tb, Zb, Yb, -0.5f, 1.5f, lane, wave);  // T = 1.5I - 0.5 Z@Y
      wg_gemm64(Ub, Yb, Tb,  1.0f, 0.0f, lane, wave);  // U = Y@T   (new Y)
      wg_gemm64(Yb, Tb, Zb,  1.0f, 0.0f, lane, wave);  // Ybuf = T@Z (new Z)
      float* nY = Ub; float* nZ = Yb; float* nU = Zb;  // rotate buffers
      Yb = nY; Zb = nZ; Ub = nU;                       // Tb stays scratch
    }
  }

  // ---- E = R - I (in place) ; S = c_K * I ----
  for (int i = tid; i < NMAT * NMAT; i += 128) {
    const int r = i >> 6, c = i & 63;
    if (r == c) Yb[r * LDA + c] -= 1.0f;
  }
  const float cK = ((NTERMS & 1) ? 1.0f : -1.0f) / (float)NTERMS;
  for (int i = tid; i < NMAT * NMAT; i += 128) {
    const int r = i >> 6, c = i & 63;
    Tb[r * LDA + c] = (r == c) ? cK : 0.0f;
  }
  __syncthreads();

  // ---- Horner: S = c_k I + E@S,  k = K-1 .. 1 ----
  float* Sb = Tb;
  float* Wb = Ub;
  for (int k = NTERMS - 1; k >= 1; --k) {
    const float ck = ((k & 1) ? 1.0f : -1.0f) / (float)k;
    wg_gemm64(Wb, Yb, Sb, 1.0f, ck, lane, wave);
    float* t = Sb; Sb = Wb; Wb = t;
  }
  // ---- log(X) = 2^s * (E@S) + ln(c) * I ----
  wg_gemm64(Wb, Yb, Sb, (float)(1 << NSQRT), log_c, lane, wave);

  // ---- write upper-triangular (row-major triu order) ----
  float* L = Wb;
  float* O = out + (size_t)blockIdx.x * TRI;
  for (int idx = tid; idx < TRI; idx += 128) {
    int r = 0, base = 0;
    while (idx >= base + (NMAT - r)) { base += NMAT - r; ++r; }
    const int c = r + (idx - base);
    O[idx] = L[r * LDA + c];
  }
}

extern "C" void kernel_launch(void* const* d_in, const int* in_sizes, int n_in,
                              void* d_out, int out_size, void* d_ws, size_t ws_size,
                              hipStream_t stream) {
  const float* x = (const float*)d_in[0];
  float* out = (float*)d_out;
  const int B = in_sizes[0] / (NMAT * NMAT);
  spd_log_triu_kernel<<<B, 128, 0, stream>>>(x, out);
}